// Huber_37718402793523
// MI455X (gfx1250) — compile-verified
//
#include <hip/hip_runtime.h>
#include <math.h>

typedef float v2f __attribute__((ext_vector_type(2)));
typedef float v8f __attribute__((ext_vector_type(8)));

#define Mdim 1024
#define Ndim 1024
#define Rdim 64
#define HITERS 3

// alpha(c) = c*(1-chi2cdf(2c,1)) + 0.5*chi2cdf(c^2,3)
// chi2cdf(x,1)=erf(sqrt(x/2)); chi2cdf(x,3)=erf(sqrt(y)) - (2/sqrt(pi))*sqrt(y)*exp(-y), y=x/2
__device__ __forceinline__ float alpha_from_c(float c) {
  float cdf1 = erff(sqrtf(c));
  float s = c * 0.7071067811865475f;                 // sqrt(c^2/2)
  float cdf3 = erff(s) - 1.1283791670955126f * s * expf(-s * s);
  return c * (1.0f - cdf1) + 0.5f * cdf3;
}

__device__ __forceinline__ float psi_clip(float x, float c) {
  return fminf(fmaxf(x, -c), c);
}

// ---------------------------------------------------------------------------
// Phase 1a: per-column masked Gram G_j = U^T diag(m_j) U via fp32 WMMA,
// in-block Cholesky, store L factor + observation count. 1 block per column.
// LDS chunk is staged K-contiguous (transposed) so each WMMA operand is a
// single 8-byte-aligned ds_load_b64 into an even VGPR pair (no repack movs).
// ---------------------------------------------------------------------------
__global__ __launch_bounds__(128)
void gram_cols(const float* __restrict__ X, const float* __restrict__ U,
               float* __restrict__ Lws, float* __restrict__ kobs_ws) {
  __shared__ float xmask[Mdim];
  __shared__ float UsT[Rdim][34];   // [column][k within 32-chunk], pad->8B align
  __shared__ float G[Rdim][Rdim + 1];
  __shared__ float red[128];

  const int j    = blockIdx.x;
  const int tid  = threadIdx.x;
  const int lane = tid & 31;
  const int wv   = tid >> 5;      // tile-row ti = wave id (0..3)
  const int half = lane >> 4;     // K-half select per ISA A/B layout
  const int mrow = lane & 15;

  float cnt = 0.f;
  for (int k = tid; k < Mdim; k += 128) {
    float x = X[k * Ndim + j];
    float m = (x != 0.0f) ? 1.0f : 0.0f;
    xmask[k] = m;
    cnt += m;
  }
  red[tid] = cnt;
  __syncthreads();
  if (tid == 0) {
    float s = 0.f;
    for (int t = 0; t < 128; ++t) s += red[t];
    kobs_ws[j] = s;
  }

  v8f acc[4] = {};
  for (int k0 = 0; k0 < Mdim; k0 += 32) {
    __syncthreads();
    for (int idx = tid; idx < 32 * Rdim; idx += 128) {
      int kk = idx >> 6, c = idx & 63;
      UsT[c][kk] = U[(k0 + kk) * Rdim + c] * xmask[k0 + kk];
    }
    __syncthreads();
    for (int kk = 0; kk < 32; kk += 4) {
      const int ka = kk + 2 * half;   // even -> 8B aligned with pitch 34
      const v2f a = *reinterpret_cast<const v2f*>(&UsT[wv * 16 + mrow][ka]);
#pragma unroll
      for (int tj = 0; tj < 4; ++tj) {
        const v2f b = *reinterpret_cast<const v2f*>(&UsT[tj * 16 + mrow][ka]);
        acc[tj] = __builtin_amdgcn_wmma_f32_16x16x4_f32(
            false, a, false, b, (short)0, acc[tj], false, false);
      }
    }
  }
  __syncthreads();
#pragma unroll
  for (int tj = 0; tj < 4; ++tj)
#pragma unroll
    for (int v = 0; v < 8; ++v)
      G[wv * 16 + v + 8 * half][tj * 16 + mrow] = acc[tj][v];
  __syncthreads();
  if (tid < Rdim) G[tid][tid] = G[tid][tid] * 1.000001f + 1e-12f;  // tiny ridge
  __syncthreads();
  // in-place Cholesky (lower)
  for (int k = 0; k < Rdim; ++k) {
    if (tid == 0) G[k][k] = sqrtf(fmaxf(G[k][k], 1e-20f));
    __syncthreads();
    float dk = G[k][k];
    for (int i = k + 1 + tid; i < Rdim; i += 128) G[i][k] /= dk;
    __syncthreads();
    for (int i = k + 1 + tid; i < Rdim; i += 128) {
      float lik = G[i][k];
      for (int jj = k + 1; jj <= i; ++jj) G[i][jj] -= lik * G[jj][k];
    }
    __syncthreads();
  }
  float* Lj = Lws + (size_t)j * (Rdim * Rdim);
  for (int idx = tid; idx < Rdim * Rdim; idx += 128) {
    int r = idx >> 6, cc = idx & 63;
    Lj[idx] = (cc <= r) ? G[r][cc] : 0.0f;
  }
}

// ---------------------------------------------------------------------------
// Phase 1b: sequential sweep over columns carrying sigma. Single block.
// ---------------------------------------------------------------------------
__global__ __launch_bounds__(256)
void col_sweep(const float* __restrict__ X, const float* __restrict__ U,
               const float* __restrict__ V, const float* __restrict__ Lws,
               const float* __restrict__ kobs_ws,
               const float* sigma_in, const float* c_in,
               const float* lamda_in, const float* mu_in,
               float* __restrict__ Vout, float* __restrict__ sigma_out) {
  __shared__ float xcol[Mdim];
  __shared__ float rr[Mdim];
  __shared__ float Lsh[Rdim][Rdim + 1];
  __shared__ float beta[Rdim], tvec[Rdim], dvec[Rdim], yvec[Rdim];
  __shared__ float part[4][Rdim];
  __shared__ float red[256];
  __shared__ float sh_sigma, sh_sig;

  const int tid = threadIdx.x;
  const float c = *c_in, lam = *lamda_in, mu = *mu_in;
  const float alpha = alpha_from_c(c);
  if (tid == 0) sh_sigma = *sigma_in;
  __syncthreads();

  for (int j = 0; j < Ndim; ++j) {
    for (int k = tid; k < Mdim; k += 256) xcol[k] = X[k * Ndim + j];
    const float* Lj = Lws + (size_t)j * (Rdim * Rdim);
    for (int idx = tid; idx < Rdim * Rdim; idx += 256)
      Lsh[idx >> 6][idx & 63] = Lj[idx];
    if (tid < Rdim) beta[tid] = V[tid * Ndim + j];
    __syncthreads();
    const float kobs = fmaxf(kobs_ws[j], 1.0f);

    for (int it = 0; it < HITERS; ++it) {
      const float sig_in = sh_sigma;
      float s1 = 0.f;
      for (int k = tid; k < Mdim; k += 256) {
        float x = xcol[k];
        float rk = 0.f;
        if (x != 0.0f) {
          float dot = 0.f;
          const float* Uk = U + k * Rdim;
          for (int cc2 = 0; cc2 < Rdim; ++cc2) dot += Uk[cc2] * beta[cc2];
          rk = x - dot;
        }
        rr[k] = rk;
        float p = psi_clip(rk / sig_in, c);
        s1 += p * p;
      }
      red[tid] = s1;
      __syncthreads();
      if (tid == 0) {
        float s = 0.f;
        for (int t = 0; t < 256; ++t) s += red[t];
        float tau = sqrtf(s / (2.0f * kobs * alpha));
        sh_sigma = tau * lam;   // persists across iterations AND columns
        sh_sig = sh_sigma;
      }
      __syncthreads();
      const float sig = sh_sig;
      {  // t = U^T (psi(r/sig)*sig)
        int cc2 = tid & 63, q = tid >> 6;
        float acc = 0.f;
        for (int k = q * 256; k < q * 256 + 256; ++k) {
          float w = psi_clip(rr[k] / sig, c) * sig;
          acc += U[k * Rdim + cc2] * w;
        }
        part[q][cc2] = acc;
      }
      __syncthreads();
      if (tid < Rdim)
        tvec[tid] = part[0][tid] + part[1][tid] + part[2][tid] + part[3][tid];
      __syncthreads();
      if (tid == 0) {  // solve L L^T delta = t
        for (int i = 0; i < Rdim; ++i) {
          float s = tvec[i];
          for (int jj = 0; jj < i; ++jj) s -= Lsh[i][jj] * yvec[jj];
          yvec[i] = s / Lsh[i][i];
        }
        for (int i = Rdim - 1; i >= 0; --i) {
          float s = yvec[i];
          for (int jj = i + 1; jj < Rdim; ++jj) s -= Lsh[jj][i] * dvec[jj];
          dvec[i] = s / Lsh[i][i];
        }
      }
      __syncthreads();
      if (tid < Rdim) beta[tid] += mu * dvec[tid];
      __syncthreads();
    }
    if (tid < Rdim) Vout[tid * Ndim + j] = beta[tid];
    __syncthreads();
  }
  if (tid == 0) *sigma_out = sh_sigma;
}

// ---------------------------------------------------------------------------
// Phase 2: independent row updates, fused Gram(WMMA)+Cholesky+iterations.
// One block per row. Vs is K-contiguous with even pitch -> direct v2f loads.
// ---------------------------------------------------------------------------
__global__ __launch_bounds__(128)
void row_update(const float* __restrict__ X, const float* __restrict__ Uin,
                const float* __restrict__ Vn, const float* sigma_f_p,
                const float* c_in, const float* lamda_in, const float* mu_in,
                float* __restrict__ Uout) {
  __shared__ float xr[Ndim];
  __shared__ float rr[Ndim];
  __shared__ float Vs[Rdim][34];    // pad 34 -> every even ka is 8B aligned
  __shared__ float G[Rdim][Rdim + 1];
  __shared__ float part[2][Rdim];
  __shared__ float red[128];
  __shared__ float beta[Rdim], tvec[Rdim], dvec[Rdim], yvec[Rdim];
  __shared__ float sh_sig;

  const int i    = blockIdx.x;
  const int tid  = threadIdx.x;
  const int lane = tid & 31;
  const int wv   = tid >> 5;
  const int half = lane >> 4;
  const int mrow = lane & 15;

  const float c = *c_in, lam = *lamda_in, mu = *mu_in;
  const float sigf = *sigma_f_p;
  const float alpha = alpha_from_c(c);

  for (int n = tid; n < Ndim; n += 128) xr[n] = X[(size_t)i * Ndim + n];
  __syncthreads();

  v8f acc[4] = {};
  for (int k0 = 0; k0 < Ndim; k0 += 32) {
    __syncthreads();
    for (int idx = tid; idx < Rdim * 32; idx += 128) {
      int cc2 = idx >> 5, nn = idx & 31;
      float m = (xr[k0 + nn] != 0.0f) ? 1.0f : 0.0f;
      Vs[cc2][nn] = Vn[cc2 * Ndim + k0 + nn] * m;
    }
    __syncthreads();
    for (int kk = 0; kk < 32; kk += 4) {
      const int ka = kk + 2 * half;
      const v2f a = *reinterpret_cast<const v2f*>(&Vs[wv * 16 + mrow][ka]);
#pragma unroll
      for (int tj = 0; tj < 4; ++tj) {
        const v2f b = *reinterpret_cast<const v2f*>(&Vs[tj * 16 + mrow][ka]);
        acc[tj] = __builtin_amdgcn_wmma_f32_16x16x4_f32(
            false, a, false, b, (short)0, acc[tj], false, false);
      }
    }
  }
  __syncthreads();
#pragma unroll
  for (int tj = 0; tj < 4; ++tj)
#pragma unroll
    for (int v = 0; v < 8; ++v)
      G[wv * 16 + v + 8 * half][tj * 16 + mrow] = acc[tj][v];
  __syncthreads();
  if (tid < Rdim) G[tid][tid] = G[tid][tid] * 1.000001f + 1e-12f;
  __syncthreads();
  for (int k = 0; k < Rdim; ++k) {
    if (tid == 0) G[k][k] = sqrtf(fmaxf(G[k][k], 1e-20f));
    __syncthreads();
    float dk = G[k][k];
    for (int ii = k + 1 + tid; ii < Rdim; ii += 128) G[ii][k] /= dk;
    __syncthreads();
    for (int ii = k + 1 + tid; ii < Rdim; ii += 128) {
      float lik = G[ii][k];
      for (int jj = k + 1; jj <= ii; ++jj) G[ii][jj] -= lik * G[jj][k];
    }
    __syncthreads();
  }

  if (tid < Rdim) beta[tid] = Uin[(size_t)i * Rdim + tid];
  __syncthreads();

  for (int it = 0; it < HITERS; ++it) {
    float s1 = 0.f;
    for (int n = tid; n < Ndim; n += 128) {
      float x = xr[n];
      float rn = 0.f;
      if (x != 0.0f) {
        float dot = 0.f;
        for (int cc2 = 0; cc2 < Rdim; ++cc2) dot += beta[cc2] * Vn[cc2 * Ndim + n];
        rn = x - dot;
      }
      rr[n] = rn;
      float p = psi_clip(rn / sigf, c);   // tau uses sigma_f (reference quirk)
      s1 += p * p;
    }
    red[tid] = s1;
    __syncthreads();
    if (tid == 0) {
      float s = 0.f;
      for (int t = 0; t < 128; ++t) s += red[t];
      float tau = sqrtf(s / (2.0f * alpha));  // len(y)==1 quirk in reference
      sh_sig = tau * lam;
    }
    __syncthreads();
    const float sig_loc = sh_sig;
    {
      int cc2 = tid & 63, hq = tid >> 6;
      float acc2 = 0.f;
      for (int n = hq * 512; n < hq * 512 + 512; ++n) {
        float w = psi_clip(rr[n] / sig_loc, c) * sigf;
        acc2 += Vn[cc2 * Ndim + n] * w;
      }
      part[hq][cc2] = acc2;
    }
    __syncthreads();
    if (tid < Rdim) tvec[tid] = part[0][tid] + part[1][tid];
    __syncthreads();
    if (tid == 0) {
      for (int ii2 = 0; ii2 < Rdim; ++ii2) {
        float s = tvec[ii2];
        for (int jj = 0; jj < ii2; ++jj) s -= G[ii2][jj] * yvec[jj];
        yvec[ii2] = s / G[ii2][ii2];
      }
      for (int ii2 = Rdim - 1; ii2 >= 0; --ii2) {
        float s = yvec[ii2];
        for (int jj = ii2 + 1; jj < Rdim; ++jj) s -= G[jj][ii2] * dvec[jj];
        dvec[ii2] = s / G[ii2][ii2];
      }
    }
    __syncthreads();
    if (tid < Rdim) beta[tid] += mu * dvec[tid];
    __syncthreads();
  }
  if (tid < Rdim) Uout[(size_t)i * Rdim + tid] = beta[tid];
}

// ---------------------------------------------------------------------------
extern "C" void kernel_launch(void* const* d_in, const int* in_sizes, int n_in,
                              void* d_out, int out_size, void* d_ws, size_t ws_size,
                              hipStream_t stream) {
  (void)in_sizes; (void)n_in; (void)out_size; (void)ws_size;
  const float* X     = (const float*)d_in[0];
  const float* U     = (const float*)d_in[1];
  const float* V     = (const float*)d_in[2];
  const float* sigma = (const float*)d_in[3];
  const float* c     = (const float*)d_in[4];
  const float* lamda = (const float*)d_in[5];
  const float* mu    = (const float*)d_in[6];

  float* out  = (float*)d_out;
  float* outX = out;                         // 1024*1024
  float* outU = out + Mdim * Ndim;           // 1024*64
  float* outV = outU + Mdim * Rdim;          // 64*1024

  float* wsf     = (float*)d_ws;
  float* sigma_f = wsf;                      // [1]
  float* kobs    = wsf + 64;                 // [1024]
  float* Lws     = wsf + 2048;               // [1024 * 64*64] = 16.8 MB

  // Output X is the unchanged input X.
  hipMemcpyAsync(outX, X, sizeof(float) * Mdim * Ndim,
                 hipMemcpyDeviceToDevice, stream);

  gram_cols<<<Ndim, 128, 0, stream>>>(X, U, Lws, kobs);
  col_sweep<<<1, 256, 0, stream>>>(X, U, V, Lws, kobs, sigma, c, lamda, mu,
                                   outV, sigma_f);
  row_update<<<Mdim, 128, 0, stream>>>(X, U, outV, sigma_f, c, lamda, mu, outU);
}